// My_loss_45337674776928
// MI455X (gfx1250) — compile-verified
//
#include <hip/hip_runtime.h>
#include <hip/hip_bf16.h>
#include <math.h>

// ---- physics constants (match reference) ----
#define RHO_L_ATM    851.0f
#define BETA_L_ATM   1.8e9f
#define AIR_FRACTION 0.005f
#define RHO_G_ATM    1.225f
#define POLY_N       1.4f
#define P_ATM        101325.0f
#define P_CRIT       3.0e6f
#define P_MIN        1.0e3f
#define V_VOL        1.0e-3f
#define EPS_DPDT     1e-12f

typedef __attribute__((ext_vector_type(2))) float v2f;
typedef __attribute__((ext_vector_type(8))) float v8f;

// Per-element masked |physics| contribution, all f32 like the reference.
__device__ __forceinline__ float elem_contrib(float outp, float dp, float md) {
    float p      = outp * 0.1f;
    float p_used = fmaxf(p, P_MIN);

    const float L     = P_CRIT - P_ATM;
    const float invL  = 1.0f / L;
    const float invL3 = invL * invL * invL;

    float x  = (p_used - P_ATM) * invL;
    float x2 = x * x;
    float theta_mid = 1.0f - 3.0f * x2 + 2.0f * x2 * x;   // smoothstep
    float theta = (p_used <= P_ATM) ? 1.0f
                : ((p_used >= P_CRIT) ? 0.0f : theta_mid);
    float dtheta_dp = ((p_used > P_ATM) && (p_used < P_CRIT))
        ? 6.0f * ((p_used - P_ATM) * (p_used - P_CRIT)) * invL3
        : 0.0f;

    const float af_ratio = AIR_FRACTION / (1.0f - AIR_FRACTION);
    float pow_term = powf(P_ATM / p_used, 1.0f / POLY_N);
    float p_denom  = af_ratio * pow_term * theta;
    float p_ratio  = af_ratio * pow_term * (theta / (p_used * POLY_N) - dtheta_dp);

    float exp_term = expf(-(p_used - P_ATM) * (1.0f / BETA_L_ATM)) * (1.0f / BETA_L_ATM);
    const float rho_mix_init = RHO_L_ATM + RHO_G_ATM * af_ratio;
    float denom = BETA_L_ATM * exp_term + p_denom;
    float drho  = rho_mix_init * (exp_term + p_ratio) / (denom * denom);

    float physics = V_VOL * drho * dp - md;
    return (fabsf(dp) >= EPS_DPDT) ? fabsf(physics) : 0.0f;
}

// Wave32 all-lane sum via V_WMMA_F32_16X16X4_F32.
// A(16x4): lane L<16 holds {s_L, 0} (K=0,1); lane L>=16 holds {s_L, 0} (K=2,3).
// B(4x16) = ones  =>  D[m][n] = s_m + s_{m+16} for every n.
// Per-lane sum of the 8 D VGPRs gives half-sums; xor-16 shuffle completes it.
__device__ __forceinline__ float wave_reduce_sum(float v) {
#if __has_builtin(__builtin_amdgcn_wmma_f32_16x16x4_f32)
    v2f a; a[0] = v;    a[1] = 0.0f;
    v2f b; b[0] = 1.0f; b[1] = 1.0f;
    v8f c = {};
    v8f d = __builtin_amdgcn_wmma_f32_16x16x4_f32(
        /*neg_a=*/false, a, /*neg_b=*/false, b,
        /*c_mod=*/(short)0, c, /*reuse_a=*/false, /*reuse_b=*/false);
    float r = ((d[0] + d[1]) + (d[2] + d[3])) + ((d[4] + d[5]) + (d[6] + d[7]));
    r += __shfl_xor(r, 16, 32);
    return r;
#else
    for (int off = 16; off > 0; off >>= 1) v += __shfl_xor(v, off, 32);
    return v;
#endif
}

__global__ __launch_bounds__(256) void physics_loss_partial(
    const float* __restrict__ outP, const float* __restrict__ dpdt,
    const float* __restrict__ mdot, float* __restrict__ partial, int n) {
    const int tid    = threadIdx.x;
    const int gid    = blockIdx.x * blockDim.x + tid;
    const int stride = gridDim.x * blockDim.x;

    float acc = 0.0f;
    const int nvec = n >> 2;
    const float4* outP4 = (const float4*)outP;
    const float4* dpdt4 = (const float4*)dpdt;
    const float4* mdot4 = (const float4*)mdot;

    for (int i = gid; i < nvec; i += stride) {
        float4 p4 = outP4[i];
        float4 d4 = dpdt4[i];
        float4 m4 = mdot4[i];
        acc += elem_contrib(p4.x, d4.x, m4.x);
        acc += elem_contrib(p4.y, d4.y, m4.y);
        acc += elem_contrib(p4.z, d4.z, m4.z);
        acc += elem_contrib(p4.w, d4.w, m4.w);
    }
    for (int i = (nvec << 2) + gid; i < n; i += stride)
        acc += elem_contrib(outP[i], dpdt[i], mdot[i]);

    float wsum = wave_reduce_sum(acc);   // EXEC all-ones here: no divergence

    __shared__ float sh[8];
    const int wave = tid >> 5;
    const int lane = tid & 31;
    if (lane == 0) sh[wave] = wsum;
    __syncthreads();
    if (tid == 0) {
        float s = 0.0f;
        const int nw = blockDim.x >> 5;
        for (int w = 0; w < nw; ++w) s += sh[w];   // fixed order: deterministic
        partial[blockIdx.x] = s;
    }
}

__global__ __launch_bounds__(256) void physics_loss_final(
    const float* __restrict__ partial, int nb,
    const float* __restrict__ log_lambda,
    float* __restrict__ out, float inv_n) {
    const int tid = threadIdx.x;
    float acc = 0.0f;
    for (int i = tid; i < nb; i += blockDim.x) acc += partial[i];

    float wsum = wave_reduce_sum(acc);

    __shared__ float sh[8];
    const int wave = tid >> 5;
    const int lane = tid & 31;
    if (lane == 0) sh[wave] = wsum;
    __syncthreads();
    if (tid == 0) {
        float s = 0.0f;
        const int nw = blockDim.x >> 5;
        for (int w = 0; w < nw; ++w) s += sh[w];
        out[0] = expf(log_lambda[0]) * s * inv_n;
    }
}

extern "C" void kernel_launch(void* const* d_in, const int* in_sizes, int n_in,
                              void* d_out, int out_size, void* d_ws, size_t ws_size,
                              hipStream_t stream) {
    (void)n_in; (void)out_size;
    // setup_inputs order: targets_P, dpdt, mdot_A, outputs_P, log_lambda
    const float* dpdt       = (const float*)d_in[1];
    const float* mdot       = (const float*)d_in[2];
    const float* outP       = (const float*)d_in[3];
    const float* log_lambda = (const float*)d_in[4];
    const int n = in_sizes[0];

    float* partial = (float*)d_ws;
    int nb = 2048;                                   // 2048 blocks x 256 thr = 8 waves/block
    if (ws_size < (size_t)nb * sizeof(float)) {
        nb = (int)(ws_size / sizeof(float));
        if (nb < 1) nb = 1;
    }

    physics_loss_partial<<<nb, 256, 0, stream>>>(outP, dpdt, mdot, partial, n);
    physics_loss_final<<<1, 256, 0, stream>>>(partial, nb, log_lambda,
                                              (float*)d_out, 1.0f / (float)n);
}